// WaveCell_15066745274910
// MI455X (gfx1250) — compile-verified
//
#include <hip/hip_runtime.h>
#include <math.h>

// ---------------------------------------------------------------------------
// WaveCell nonlinear FDTD scan for MI455X (gfx1250).
//   * whole 256x256 field (+zero halo) resident in one WGP's 320KB LDS
//   * 1 workgroup per batch (grid=4), 1024 threads = 32 wave32 waves
//   * each thread owns 64 vertically-contiguous rows of one column, so
//     north/south stencil neighbors live in VGPRs (register chain);
//     only east/west come from LDS (1 ds_load_2addr per cell)
//   * y2 / y1-center time history held in per-thread VGPRs
//   * per-cell constants folded offline into float4, streamed from L2 (b128)
//   * CDNA5 paths: global_load_async_to_lds_b32 + s_wait_asynccnt,
//     global_prefetch_b8, split workgroup barriers
// ---------------------------------------------------------------------------

#define NXg      256
#define NYg      256
#define TSTEPS   256
#define LPITCH   258          // 256 + zero halo on both sides

#define C0c      1.0f
#define C1c      0.9f
#define ETAc     0.5f
#define BETAc    100.0f
#define NLCc     0.1f

// Generic-pointer low 32 bits == raw LDS offset (flat aperture mapping).
__device__ __forceinline__ unsigned lds_offset(const void* p) {
  return (unsigned)(unsigned long long)p;
}

// ---------------------------------------------------------------------------
// One-shot: proj = smoothed+projected rho; fold per-cell constants:
//   K = { C0+(C1-C0)*proj,  NL_C*proj,  proj,  b_boundary }
// Stencil rows are staged global->LDS with CDNA5 async LDS loads.
// ---------------------------------------------------------------------------
__global__ void wave_precompute(const float* __restrict__ rho,
                                const float* __restrict__ bbound,
                                float4* __restrict__ K) {
  __shared__ float rbuf[3][LPITCH];
  const int r = blockIdx.x;        // one row per block
  const int c = threadIdx.x;       // 256 threads

  // zero halo + rows that may be skipped at the grid edge
  for (int i = threadIdx.x; i < 3 * LPITCH; i += blockDim.x)
    ((float*)rbuf)[i] = 0.0f;
  __syncthreads();

  // async copy of the three stencil rows of rho into LDS (ASYNCcnt path)
#pragma unroll
  for (int dr = -1; dr <= 1; ++dr) {
    const int rr = r + dr;
    if (rr >= 0 && rr < NXg) {
      const float* gsrc = rho + rr * NYg + c;
      unsigned ldst = lds_offset(&rbuf[dr + 1][c + 1]);
      asm volatile("global_load_async_to_lds_b32 %0, %1, off"
                   :: "v"(ldst), "v"(gsrc) : "memory");
    }
  }
  asm volatile("s_wait_asynccnt 0x0" ::: "memory");
  __syncthreads();

  const float lpf = 0.5f * rbuf[1][c + 1]
                  + 0.125f * (rbuf[0][c + 1] + rbuf[2][c + 1] +
                              rbuf[1][c]     + rbuf[1][c + 2]);
  const float t0   = tanhf(BETAc * ETAc);
  const float den  = t0 + tanhf(BETAc * (1.0f - ETAc));
  const float proj = (t0 + tanhf(BETAc * (lpf - ETAc))) / den;

  const int gi = r * NYg + c;
  K[gi] = make_float4(C0c + (C1c - C0c) * proj,   // c_lin
                      NLCc * proj,                // nonlinear speed coeff
                      proj,                       // nonlinear damping coeff
                      bbound[gi]);                // PML damping
}

// ---------------------------------------------------------------------------
// 256-step scan. grid = 4 (batch), block = 1024.  Dynamic LDS:
//   [0 .. 258*258)  padded y1 field (halo stays zero forever)
//   [258*258 .. +8) probe reduction slots
// Thread t owns column (t&255), rows r = 64*(t>>8) + k, k = 0..63
// (vertically contiguous -> N/S neighbors are registers).
// ---------------------------------------------------------------------------
__global__ void __launch_bounds__(1024, 1)
wave_scan(const float* __restrict__ x,
          const float4* __restrict__ K,
          float* __restrict__ out) {
  extern __shared__ float lds[];
  float* red = lds + LPITCH * LPITCH;

  const int b    = blockIdx.x;
  const int tid  = threadIdx.x;
  const int col  = tid & 255;
  const int rg   = tid >> 8;
  const int rbase = rg << 6;                 // first owned row

  for (int i = tid; i < LPITCH * LPITCH + 8; i += 1024) lds[i] = 0.0f;

  float yc[64];   // y1 at own cells (old value during phase A)
  float y2[64];   // y2 at own cells
#pragma unroll
  for (int k = 0; k < 64; ++k) { yc[k] = 0.0f; y2[k] = 0.0f; }

  // warm the L2-resident constant table (global_prefetch_b8)
#pragma unroll
  for (int k = 0; k < 64; ++k)
    __builtin_prefetch(&K[(rbase + k) * NYg + col], 0, 3);

  __syncthreads();

  const float Hc     = 1.41421356237309515f * 1.01f;  // DT*max(C0,C1)*sqrt2*1.01
  const float inv_h2 = 1.0f / (Hc * Hc);
  float Iacc = 0.0f;                                   // probe accumulator

#pragma unroll 1
  for (int t = 0; t < TSTEPS; ++t) {
    const float xsrc = x[b * TSTEPS + t];              // uniform -> s_load

    // ----- phase A: compute y_new from old field -----
    // North neighbor of row rbase (owned by rg-1, or zero halo): one LDS read.
    float prev = lds[rbase * LPITCH + (col + 1)];

#pragma unroll
    for (int k = 0; k < 64; ++k) {
      const int   r   = rbase + k;
      const float4 Kc = K[r * NYg + col];              // global b128, L2 hit
      const float uc  = yc[k];
      const int   li  = (r + 1) * LPITCH + (col + 1);

      const float un  = prev;                           // register chain
      const float us  = (k < 63) ? yc[k + 1]            // register (old value)
                                 : lds[li + LPITCH];    // inter-group boundary
      const float uw  = lds[li - 1];                    // ds_load_2addr pair
      const float ue  = lds[li + 1];

      const float u2  = uc * uc;
      const float bp  = Kc.w + Kc.z * (1.0f / (1.0f + u2));   // damping
      const float cc  = Kc.x + Kc.y * u2;                     // wave speed
      const float lap = (un + us + uw + ue - 4.0f * uc) * inv_h2;

      float yn = (2.0f * uc - (1.0f - 0.5f * bp) * y2[k] + cc * cc * lap)
               * (1.0f / (1.0f + 0.5f * bp));

      if (k == 40 && tid == 128) yn += xsrc;            // source (40,128): rg0,k40
      if (k == 24 && (tid == 768 + 90 || tid == 768 + 128 || tid == 768 + 166))
        Iacc += yn * yn;                                // probes (216,{90,128,166})

      prev  = uc;                                       // old y1 for next row
      y2[k] = uc;                                       // y2 <- old y1
      yc[k] = yn;                                       // y1 <- y_new (regs)
    }
    __syncthreads();   // all reads of the old field are done

    // ----- phase B: commit new field to LDS -----
#pragma unroll
    for (int k = 0; k < 64; ++k)
      lds[(rbase + k + 1) * LPITCH + (col + 1)] = yc[k];
    __syncthreads();   // new field visible before next step
  }

  if (tid == 768 + 90)  red[0] = Iacc;
  if (tid == 768 + 128) red[1] = Iacc;
  if (tid == 768 + 166) red[2] = Iacc;
  __syncthreads();
  if (tid == 0) {
    const float inv = 1.0f / (red[0] + red[1] + red[2]);
    out[b * 3 + 0] = red[0] * inv;
    out[b * 3 + 1] = red[1] * inv;
    out[b * 3 + 2] = red[2] * inv;
  }
}

// ---------------------------------------------------------------------------
extern "C" void kernel_launch(void* const* d_in, const int* in_sizes, int n_in,
                              void* d_out, int out_size, void* d_ws, size_t ws_size,
                              hipStream_t stream) {
  (void)in_sizes; (void)n_in; (void)out_size; (void)ws_size;
  const float* x      = (const float*)d_in[0];   // (4,256)
  const float* rho    = (const float*)d_in[1];   // (256,256)
  const float* bbound = (const float*)d_in[2];   // (256,256)
  float4*      Kt     = (float4*)d_ws;           // 65536 * 16B = 1 MB scratch
  float*       out    = (float*)d_out;           // (4,3)

  wave_precompute<<<dim3(NXg), dim3(NYg), 0, stream>>>(rho, bbound, Kt);

  const size_t shbytes = (size_t)(LPITCH * LPITCH + 8) * sizeof(float);
  wave_scan<<<dim3(4), dim3(1024), shbytes, stream>>>(x, Kt, out);
}